// TopK_64407329571091
// MI455X (gfx1250) — compile-verified
//
#include <hip/hip_runtime.h>
#include <stdint.h>

// Problem constants (from reference): B=4096 rows, D=65536, k=64.
#define ROW_LEN   65536
#define TOPK      64
#define NTHREADS  1024
#define VEC_PER_T (ROW_LEN / NTHREADS / 4)   // 16 float4 per thread
#define NREP      16                          // replicated histograms
#define REPSTRIDE 257                         // pad to spread replicas across banks
#define HISTW     (NREP * REPSTRIDE)
#define CAND_CAP  4096

// Monotonic float->uint key: larger key <=> larger float.
__device__ __forceinline__ uint32_t f2key(uint32_t b) {
  return (b & 0x80000000u) ? ~b : (b | 0x80000000u);
}
__device__ __forceinline__ float key2float(uint32_t k) {
  return __uint_as_float((k & 0x80000000u) ? (k ^ 0x80000000u) : ~k);
}

// Wave-aggregated append: one counter atomic per wave ballot instead of per lane.
__device__ __forceinline__ uint32_t wave_append(uint32_t* counter, bool pred) {
  const uint32_t mask = (uint32_t)__ballot(pred);   // wave32: low 32 bits
  if (mask == 0u) return 0xFFFFFFFFu;
  const int lane = (int)(threadIdx.x & 31u);
  const int leader = __ffs(mask) - 1;
  uint32_t base = 0u;
  if (lane == leader) base = atomicAdd(counter, (uint32_t)__popc(mask));
  base = __shfl(base, leader, 32);
  if (!pred) return 0xFFFFFFFFu;
  return base + (uint32_t)__popc(mask & ((1u << lane) - 1u));
}

__global__ __launch_bounds__(NTHREADS)
void topk_scatter_kernel(const float* __restrict__ x, float* __restrict__ out) {
  // 256 KB row buffer + ~50 KB select state: fits in CDNA5's 320 KB LDS/WGP.
  __shared__ __align__(16) uint32_t s_row[ROW_LEN];
  __shared__ uint32_t s_hist[HISTW];       // 16 replicated 256-bin histograms
  __shared__ uint32_t s_histS[256];        // reduced histogram
  __shared__ uint32_t s_candKey[CAND_CAP]; // compacted pass-1 survivors
  __shared__ uint32_t s_candIdx[CAND_CAP];
  __shared__ uint32_t s_sel[TOPK];
  __shared__ uint32_t s_bc[4];             // 0:prefix 1:remaining 2:candCnt 3:eqCnt

  const int tid = threadIdx.x;
  const float* rp = x + (size_t)blockIdx.x * ROW_LEN;
  float*       op = out + (size_t)blockIdx.x * ROW_LEN;
  const uint32_t rbase = (uint32_t)(tid & (NREP - 1)) * REPSTRIDE;

  // ---- Issue ALL async global->LDS b128 copies for this row (ASYNCcnt) ----
  const uint64_t gbase = (uint64_t)(uintptr_t)rp;
  #pragma unroll
  for (int v = 0; v < VEC_PER_T; ++v) {
    const int idx = (v * NTHREADS + tid) * 4;
    const uint32_t lds_addr = (uint32_t)(uintptr_t)(&s_row[idx]);
    const uint64_t g_addr   = gbase + (uint64_t)idx * 4u;
    asm volatile("global_load_async_to_lds_b128 %0, %1, off"
                 :: "v"(lds_addr), "v"(g_addr) : "memory");
  }

  // Zero pass-0 histograms / broadcast state while the DMA is in flight.
  for (int i = tid; i < HISTW; i += NTHREADS) s_hist[i] = 0u;
  if (tid == 0) { s_bc[0] = 0u; s_bc[1] = TOPK; s_bc[2] = 0u; s_bc[3] = 0u; }
  __syncthreads();

  // Bin-scan helper: reduce replicas, pick bin from the top, extend prefix.
  auto finish_pass = [&](int shift) {
    __syncthreads();
    if (tid < 256) {
      uint32_t s = 0u;
      #pragma unroll
      for (int r = 0; r < NREP; ++r) s += s_hist[r * REPSTRIDE + tid];
      s_histS[tid] = s;
    }
    __syncthreads();
    if (tid == 0) {
      uint32_t rem = s_bc[1], cum = 0u; int b = 255;
      for (; b > 0; --b) { const uint32_t c = s_histS[b]; if (cum + c >= rem) break; cum += c; }
      s_bc[1] = rem - cum;
      s_bc[0] |= ((uint32_t)b << shift);
    }
    __syncthreads();
  };

  // ---- Pass 0 (top byte), overlapped with staging: async loads complete
  // in order, and each thread only reads the chunks it staged itself. ----
  #define P0CHUNK(v) { \
    const int idx = ((v) * NTHREADS + tid) * 4; \
    const uint4 r = *(const uint4*)&s_row[idx]; \
    const uint32_t b0 = f2key(r.x) >> 24, b1 = f2key(r.y) >> 24; \
    const uint32_t b2 = f2key(r.z) >> 24, b3 = f2key(r.w) >> 24; \
    if (b0 == b1 && b0 == b2 && b0 == b3) atomicAdd(&s_hist[rbase + b0], 4u); \
    else { atomicAdd(&s_hist[rbase + b0], 1u); atomicAdd(&s_hist[rbase + b1], 1u); \
           atomicAdd(&s_hist[rbase + b2], 1u); atomicAdd(&s_hist[rbase + b3], 1u); } }

  asm volatile("s_wait_asynccnt 0xc" ::: "memory");
  P0CHUNK(0)  P0CHUNK(1)  P0CHUNK(2)  P0CHUNK(3)
  asm volatile("s_wait_asynccnt 0x8" ::: "memory");
  P0CHUNK(4)  P0CHUNK(5)  P0CHUNK(6)  P0CHUNK(7)
  asm volatile("s_wait_asynccnt 0x4" ::: "memory");
  P0CHUNK(8)  P0CHUNK(9)  P0CHUNK(10) P0CHUNK(11)
  asm volatile("s_wait_asynccnt 0x0" ::: "memory");
  P0CHUNK(12) P0CHUNK(13) P0CHUNK(14) P0CHUNK(15)
  #undef P0CHUNK
  finish_pass(24);

  // ---- Pass 1 (full-row scan) + candidate compaction (wave-aggregated) ----
  for (int i = tid; i < HISTW; i += NTHREADS) s_hist[i] = 0u;
  __syncthreads();
  {
    const uint32_t pfx = s_bc[0];
    for (int v = 0; v < VEC_PER_T; ++v) {
      const int idx = (v * NTHREADS + tid) * 4;
      const uint4 r = *(const uint4*)&s_row[idx];
      const uint32_t kk[4] = { f2key(r.x), f2key(r.y), f2key(r.z), f2key(r.w) };
      #pragma unroll
      for (int j = 0; j < 4; ++j) {
        const bool m = (kk[j] >> 24) == (pfx >> 24);
        if (m) atomicAdd(&s_hist[rbase + ((kk[j] >> 16) & 0xFFu)], 1u);
        const uint32_t slot = wave_append(&s_bc[2], m);
        if (m && slot < CAND_CAP) { s_candKey[slot] = kk[j]; s_candIdx[slot] = (uint32_t)(idx + j); }
      }
    }
  }
  finish_pass(16);

  const uint32_t candTotal = s_bc[2];
  const uint32_t candN = candTotal < CAND_CAP ? candTotal : (uint32_t)CAND_CAP;
  const bool overflow = candTotal > CAND_CAP;   // fallback: rescan full row

  // ---- Passes 2 & 3: scan only candidates (or full row on overflow) ----
  for (int p = 2; p <= 3; ++p) {
    const int shift = (p == 2) ? 8 : 0;
    for (int i = tid; i < HISTW; i += NTHREADS) s_hist[i] = 0u;
    __syncthreads();
    const uint32_t pfx = s_bc[0];
    if (!overflow) {
      for (uint32_t e = tid; e < candN; e += NTHREADS) {
        const uint32_t k = s_candKey[e];
        const bool m = (p == 2) ? true : (((k ^ pfx) >> 8) == 0u);
        if (m) atomicAdd(&s_hist[rbase + ((k >> shift) & 0xFFu)], 1u);
      }
    } else {
      for (int v = 0; v < VEC_PER_T; ++v) {
        const int idx = (v * NTHREADS + tid) * 4;
        const uint4 r = *(const uint4*)&s_row[idx];
        const uint32_t kk[4] = { f2key(r.x), f2key(r.y), f2key(r.z), f2key(r.w) };
        #pragma unroll
        for (int j = 0; j < 4; ++j)
          if (((kk[j] ^ pfx) >> (shift + 8)) == 0u)
            atomicAdd(&s_hist[rbase + ((kk[j] >> shift) & 0xFFu)], 1u);
      }
    }
    finish_pass(shift);
  }

  const uint32_t T    = s_bc[0];   // exact key of 64th-largest element
  const uint32_t need = s_bc[1];   // how many keys == T are inside the top-k

  // ---- Equals analysis (ties): candidates contain every key == T ----
  __syncthreads();
  if (!overflow) {
    for (uint32_t e = tid; e < candN; e += NTHREADS)
      if (s_candKey[e] == T) atomicAdd(&s_bc[3], 1u);
  } else {
    for (int v = 0; v < VEC_PER_T; ++v) {
      const int idx = (v * NTHREADS + tid) * 4;
      const uint4 r = *(const uint4*)&s_row[idx];
      const uint32_t kk[4] = { f2key(r.x), f2key(r.y), f2key(r.z), f2key(r.w) };
      #pragma unroll
      for (int j = 0; j < 4; ++j)
        if (kk[j] == T) {
          const uint32_t slot = atomicAdd(&s_bc[3], 1u);
          if (slot < CAND_CAP) s_candIdx[slot] = (uint32_t)(idx + j);  // reuse as eq list
        }
    }
  }
  __syncthreads();
  const uint32_t eqCnt = s_bc[3];
  if (eqCnt > need && tid == 0) {
    // Rare tie case: pick the `need` smallest indices (matches top_k tie order).
    uint32_t last = 0u;
    for (uint32_t rsel = 0; rsel < need; ++rsel) {
      uint32_t best = 0xFFFFFFFFu;
      if (!overflow) {
        for (uint32_t e = 0; e < candN; ++e)
          if (s_candKey[e] == T) {
            const uint32_t c = s_candIdx[e];
            if ((rsel == 0 || c > last) && c < best) best = c;
          }
      } else {
        const uint32_t n = eqCnt < CAND_CAP ? eqCnt : (uint32_t)CAND_CAP;
        for (uint32_t e = 0; e < n; ++e) {
          const uint32_t c = s_candIdx[e];
          if ((rsel == 0 || c > last) && c < best) best = c;
        }
      }
      s_sel[rsel] = best; last = best;
    }
  }
  __syncthreads();

  // ---- Output pass: strictly-greater keys get relu(v); everything else 0.
  //      Keys == T are written 0 here and fixed up below. ----
  for (int v = 0; v < VEC_PER_T; ++v) {
    const int idx = (v * NTHREADS + tid) * 4;
    const uint4 r = *(const uint4*)&s_row[idx];
    const uint32_t rb[4] = { r.x, r.y, r.z, r.w };
    float of[4];
    #pragma unroll
    for (int j = 0; j < 4; ++j) {
      const uint32_t kk = f2key(rb[j]);
      const float f = __uint_as_float(rb[j]);
      of[j] = (kk > T) ? fmaxf(f, 0.0f) : 0.0f;
    }
    float4 o; o.x = of[0]; o.y = of[1]; o.z = of[2]; o.w = of[3];
    *(float4*)(op + idx) = o;
  }

  // Order the provisional 0-stores before the fix-up stores.
  asm volatile("s_wait_storecnt 0x0" ::: "memory");
  __syncthreads();

  // ---- Tiny fix-up: write relu(value)==fT at the selected equal positions ----
  const float fT = key2float(T);
  if (fT > 0.0f) {          // if fT <= 0, relu gives 0 which is already stored
    if (eqCnt <= need) {    // common case: every equal element is selected
      if (!overflow) {
        for (uint32_t e = tid; e < candN; e += NTHREADS)
          if (s_candKey[e] == T) op[s_candIdx[e]] = fT;
      } else {
        const uint32_t n = eqCnt < CAND_CAP ? eqCnt : (uint32_t)CAND_CAP;
        for (uint32_t e = tid; e < n; e += NTHREADS) op[s_candIdx[e]] = fT;
      }
    } else {
      for (uint32_t rsel = tid; rsel < need; rsel += NTHREADS) op[s_sel[rsel]] = fT;
    }
  }
}

extern "C" void kernel_launch(void* const* d_in, const int* in_sizes, int n_in,
                              void* d_out, int out_size, void* d_ws, size_t ws_size,
                              hipStream_t stream) {
  const float* x = (const float*)d_in[0];   // [B, 65536] fp32
  // d_in[1] is k (==64), baked in at compile time as TOPK.
  float* out = (float*)d_out;               // [B, 65536] fp32
  const int rows = in_sizes[0] / ROW_LEN;   // 4096
  hipLaunchKernelGGL(topk_scatter_kernel, dim3(rows), dim3(NTHREADS), 0, stream,
                     x, out);
  (void)n_in; (void)out_size; (void)d_ws; (void)ws_size;
}